// GAT_Graph_34497177322037
// MI455X (gfx1250) — compile-verified
//
#include <hip/hip_runtime.h>
#include <cmath>

// ---------------------------------------------------------------------------
// GAT (3 layers, H=4 heads, C=32) + pooling + output projection for gfx1250.
// fp32 throughout (matches reference); dense GEMMs via V_WMMA_F32_16X16X4_F32.
// ---------------------------------------------------------------------------

#define GN   50000      // nodes
#define GE   800000     // raw edges
#define GETOT (GE + GN) // edges + self loops
#define GG   256        // graphs
#define GH   4          // heads
#define GC   32         // per-head channels
#define GD   128        // feature dim = GH*GC
#define GEMB 128

typedef float v2f __attribute__((ext_vector_type(2)));
typedef float v8f __attribute__((ext_vector_type(8)));

// ---------------- utility: float atomic max via integer atomics -------------
__device__ __forceinline__ void atomicMaxFloat(float* addr, float val) {
    if (val >= 0.0f) {
        atomicMax((int*)addr, __float_as_int(val));
    } else {
        atomicMin((unsigned int*)addr, __float_as_uint(val));
    }
}

// ---------------- fill ------------------------------------------------------
__global__ void fill_f32(float* __restrict__ p, float v, int n) {
    int i = blockIdx.x * blockDim.x + threadIdx.x;
    if (i < n) p[i] = v;
}

// ---------------- WMMA f32 GEMM: C[M,N] = A[M,K] @ B[K,N] (+bias[N]) --------
// One 16x16 output tile per wave32. M%16==0, K%4==0, N%16==0 required.
// A-frag (16x4 f32, ISA 7.12.2): lanes 0-15 -> M=lane, K = k0+{0,1};
//                                lanes 16-31 -> M=lane-16, K = k0+{2,3}.
// B-frag (4x16 f32, mirrored):   VGPR0/1 hold K=k0+2*half +{0,1}, N=lane&15.
// C/D (16x16 f32): VGPR j: lanes 0-15 M=j, lanes 16-31 M=j+8, N=lane&15.
__global__ void gemm_wmma_f32(const float* __restrict__ A,
                              const float* __restrict__ B,
                              const float* __restrict__ bias,
                              float* __restrict__ C,
                              int M, int K, int N) {
    const int wave = threadIdx.x >> 5;
    const int lane = threadIdx.x & 31;
    const int tilesN = N >> 4;
    const int total  = (M >> 4) * tilesN;
    const int tile = blockIdx.x * (blockDim.x >> 5) + wave;
    if (tile >= total) return;          // uniform per-wave: EXEC stays all-ones

    const int row0 = (tile / tilesN) << 4;
    const int col0 = (tile % tilesN) << 4;
    const int half = lane >> 4;         // 0: lanes 0-15, 1: lanes 16-31
    const int l16  = lane & 15;

    const float* Ap = A + (size_t)(row0 + l16) * K + 2 * half;
    const float* Bp = B + (size_t)(2 * half) * N + col0 + l16;

    v8f acc = {};
    for (int k0 = 0; k0 < K; k0 += 4) {
        v2f a, b;
        a.x = Ap[k0];
        a.y = Ap[k0 + 1];
        b.x = Bp[(size_t)k0 * N];
        b.y = Bp[(size_t)(k0 + 1) * N];
        acc = __builtin_amdgcn_wmma_f32_16x16x4_f32(
                false, a, false, b, (short)0, acc, false, false);
    }

    const float bv = bias ? bias[col0 + l16] : 0.0f;
    float* Cp = C + (size_t)(row0 + 8 * half) * N + col0 + l16;
#pragma unroll
    for (int j = 0; j < 8; ++j) {
        Cp[(size_t)j * N] = acc[j] + bv;
    }
}

// ---------------- attention coefficients al_src/al_dst [N,H] ----------------
__global__ void attn_proj_k(const float* __restrict__ ht,
                            const float* __restrict__ asrc,
                            const float* __restrict__ adst,
                            float* __restrict__ als,
                            float* __restrict__ ald) {
    int tid = blockIdx.x * blockDim.x + threadIdx.x;
    if (tid >= GN * GH) return;
    const int n = tid >> 2;
    const int h = tid & 3;
    const float* hp = ht + (size_t)n * GD + h * GC;
    const float* ap = asrc + h * GC;
    const float* dp = adst + h * GC;
    float s = 0.0f, d = 0.0f;
#pragma unroll
    for (int c = 0; c < GC; ++c) {
        float v = hp[c];
        s += v * ap[c];
        d += v * dp[c];
    }
    als[tid] = s;
    ald[tid] = d;
}

__device__ __forceinline__ void edge_sd(const int* __restrict__ ei, int e,
                                        int& s, int& d) {
    if (e < GE) { s = ei[e]; d = ei[GE + e]; }
    else        { s = e - GE; d = e - GE; }   // self loop
}

// ---------------- pass 1: segment max of leaky_relu(e) over dst -------------
__global__ void edge_max_k(const int* __restrict__ ei,
                           const float* __restrict__ als,
                           const float* __restrict__ ald,
                           float* __restrict__ m) {
    int e = blockIdx.x * blockDim.x + threadIdx.x;
    if (e >= GETOT) return;
    int s, d;
    edge_sd(ei, e, s, d);
#pragma unroll
    for (int h = 0; h < GH; ++h) {
        float v = als[s * GH + h] + ald[d * GH + h];
        v = (v >= 0.0f) ? v : 0.2f * v;
        atomicMaxFloat(m + (size_t)d * GH + h, v);
    }
}

// ---------------- pass 2: segment sum of exp(e - m[dst]) --------------------
__global__ void edge_sum_k(const int* __restrict__ ei,
                           const float* __restrict__ als,
                           const float* __restrict__ ald,
                           const float* __restrict__ m,
                           float* __restrict__ denom) {
    int e = blockIdx.x * blockDim.x + threadIdx.x;
    if (e >= GETOT) return;
    int s, d;
    edge_sd(ei, e, s, d);
#pragma unroll
    for (int h = 0; h < GH; ++h) {
        float v = als[s * GH + h] + ald[d * GH + h];
        v = (v >= 0.0f) ? v : 0.2f * v;
        float p = expf(v - m[(size_t)d * GH + h]);
        atomicAdd(denom + (size_t)d * GH + h, p);
    }
}

// ---------------- pass 3: agg[dst] += alpha * h[src] ------------------------
__global__ void edge_agg_k(const int* __restrict__ ei,
                           const float* __restrict__ als,
                           const float* __restrict__ ald,
                           const float* __restrict__ m,
                           const float* __restrict__ denom,
                           const float* __restrict__ ht,
                           float* __restrict__ agg) {
    int tid = blockIdx.x * blockDim.x + threadIdx.x;
    if (tid >= GETOT * GH) return;
    const int e = tid >> 2;
    const int h = tid & 3;
    int s, d;
    edge_sd(ei, e, s, d);
    float v = als[s * GH + h] + ald[d * GH + h];
    v = (v >= 0.0f) ? v : 0.2f * v;
    float p = expf(v - m[(size_t)d * GH + h]);
    float alpha = p / denom[(size_t)d * GH + h];
    const float* hs = ht + (size_t)s * GD + h * GC;
    float* ag = agg + (size_t)d * GD + h * GC;
#pragma unroll
    for (int c = 0; c < GC; ++c) {
        atomicAdd(ag + c, hs[c] * alpha);
    }
}

// ---------------- bias + LayerNorm + leaky_relu(0.1), wave32/node -----------
__global__ void post_ln_k(const float* __restrict__ agg,
                          const float* __restrict__ bias,
                          const float* __restrict__ gamma,
                          const float* __restrict__ beta,
                          float* __restrict__ out) {
    const int wave = threadIdx.x >> 5;
    const int lane = threadIdx.x & 31;
    const int node = blockIdx.x * (blockDim.x >> 5) + wave;
    if (node >= GN) return;
    const float* ap = agg + (size_t)node * GD;
    float v[4];
    float s = 0.0f, sq = 0.0f;
#pragma unroll
    for (int j = 0; j < 4; ++j) {
        int f = j * 32 + lane;
        v[j] = ap[f] + bias[f];
        s  += v[j];
        sq += v[j] * v[j];
    }
#pragma unroll
    for (int off = 16; off > 0; off >>= 1) {   // wave32 butterfly
        s  += __shfl_xor(s,  off, 32);
        sq += __shfl_xor(sq, off, 32);
    }
    const float mean = s * (1.0f / GD);
    const float var  = sq * (1.0f / GD) - mean * mean;
    const float rstd = rsqrtf(var + 1e-5f);
    float* op = out + (size_t)node * GD;
#pragma unroll
    for (int j = 0; j < 4; ++j) {
        int f = j * 32 + lane;
        float y = (v[j] - mean) * rstd * gamma[f] + beta[f];
        op[f] = (y >= 0.0f) ? y : 0.1f * y;
    }
}

// ---------------- pooling ---------------------------------------------------
__global__ void pool_k(const float* __restrict__ x,
                       const int* __restrict__ batch,
                       float* __restrict__ gmax,
                       float* __restrict__ gsum,
                       float* __restrict__ cnt) {
    int tid = blockIdx.x * blockDim.x + threadIdx.x;
    if (tid >= GN * GD) return;
    const int n = tid >> 7;
    const int f = tid & 127;
    const int b = batch[n];
    const float v = x[tid];
    atomicMaxFloat(gmax + (size_t)b * GD + f, v);
    atomicAdd(gsum + (size_t)b * GD + f, v);
    if (f == 0) atomicAdd(cnt + b, 1.0f);
}

__global__ void pool_fin_k(const float* __restrict__ gmax,
                           const float* __restrict__ gsum,
                           const float* __restrict__ cnt,
                           float* __restrict__ pooled) {
    int tid = blockIdx.x * blockDim.x + threadIdx.x;
    if (tid >= GG * 2 * GD) return;
    const int g = tid >> 8;
    const int f = tid & 255;
    pooled[tid] = (f < GD) ? gmax[(size_t)g * GD + f]
                           : gsum[(size_t)g * GD + (f - GD)] / fmaxf(cnt[g], 1.0f);
}

// ---------------------------------------------------------------------------
extern "C" void kernel_launch(void* const* d_in, const int* in_sizes, int n_in,
                              void* d_out, int out_size, void* d_ws, size_t ws_size,
                              hipStream_t stream) {
    const float* x       = (const float*)d_in[0];
    const int*   ei      = (const int*)  d_in[1];
    const int*   batch   = (const int*)  d_in[2];
    const float* W       = (const float*)d_in[3];   // [L,128,128]
    const float* att_src = (const float*)d_in[4];   // [L,4,32]
    const float* att_dst = (const float*)d_in[5];
    const float* bias    = (const float*)d_in[6];   // [L,128]
    const float* ln_g    = (const float*)d_in[7];
    const float* ln_b    = (const float*)d_in[8];
    const float* W_out   = (const float*)d_in[9];   // [256,128]
    const float* b_out   = (const float*)d_in[10];  // [128]
    float* out = (float*)d_out;

    // workspace layout (floats)
    float* ws     = (float*)d_ws;
    float* ht     = ws;                          // [N,128] transformed feats
    float* agg    = ht     + (size_t)GN * GD;    // [N,128] aggregation
    float* nodeio = agg    + (size_t)GN * GD;    // [N,128] layer output
    float* als    = nodeio + (size_t)GN * GD;    // [N,4]
    float* ald    = als    + (size_t)GN * GH;    // [N,4]
    float* segm   = ald    + (size_t)GN * GH;    // [N,4] segment max
    float* denom  = segm   + (size_t)GN * GH;    // [N,4]
    float* gmax   = denom  + (size_t)GN * GH;    // [G,128]
    float* gsum   = gmax   + (size_t)GG * GD;    // [G,128]
    float* cnt    = gsum   + (size_t)GG * GD;    // [G]
    float* pooled = cnt    + GG;                 // [G,256]

    const float NEG_INF = -INFINITY;
    const int TB = 256;
    auto blocks = [](long long n, int tb) { return (int)((n + tb - 1) / tb); };

    for (int l = 0; l < 3; ++l) {
        const float* in = (l == 0) ? x : nodeio;

        // h = in @ W[l]  -> ht           (WMMA f32, 25000 tiles, 4 waves/block)
        {
            int tiles = (GN / 16) * (GD / 16);
            gemm_wmma_f32<<<blocks(tiles, 4), 128, 0, stream>>>(
                in, W + (size_t)l * GD * GD, nullptr, ht, GN, GD, GD);
        }

        attn_proj_k<<<blocks((long long)GN * GH, TB), TB, 0, stream>>>(
            ht, att_src + l * GH * GC, att_dst + l * GH * GC, als, ald);

        fill_f32<<<blocks((long long)GN * GH, TB), TB, 0, stream>>>(segm, NEG_INF, GN * GH);
        fill_f32<<<blocks((long long)GN * GH, TB), TB, 0, stream>>>(denom, 0.0f, GN * GH);
        fill_f32<<<blocks((long long)GN * GD, TB), TB, 0, stream>>>(agg, 0.0f, GN * GD);

        edge_max_k<<<blocks(GETOT, TB), TB, 0, stream>>>(ei, als, ald, segm);
        edge_sum_k<<<blocks(GETOT, TB), TB, 0, stream>>>(ei, als, ald, segm, denom);
        edge_agg_k<<<blocks((long long)GETOT * GH, TB), TB, 0, stream>>>(
            ei, als, ald, segm, denom, ht, agg);

        post_ln_k<<<blocks(GN, 8), 256, 0, stream>>>(   // 8 wave32 nodes / block
            agg, bias + l * GD, ln_g + l * GD, ln_b + l * GD, nodeio);
    }

    // pooling
    fill_f32<<<blocks(GG * GD, TB), TB, 0, stream>>>(gmax, NEG_INF, GG * GD);
    fill_f32<<<blocks(GG * GD, TB), TB, 0, stream>>>(gsum, 0.0f, GG * GD);
    fill_f32<<<blocks(GG, TB), TB, 0, stream>>>(cnt, 0.0f, GG);
    pool_k<<<blocks((long long)GN * GD, TB), TB, 0, stream>>>(nodeio, batch, gmax, gsum, cnt);
    pool_fin_k<<<blocks(GG * 2 * GD, TB), TB, 0, stream>>>(gmax, gsum, cnt, pooled);

    // out = pooled @ W_out + b_out       (WMMA f32, 128 tiles)
    {
        int tiles = (GG / 16) * (GEMB / 16);
        gemm_wmma_f32<<<blocks(tiles, 4), 128, 0, stream>>>(
            pooled, W_out, b_out, out, GG, 2 * GD, GEMB);
    }
}